// ReactionDiffusionSystem_23098334117922
// MI455X (gfx1250) — compile-verified
//
#include <hip/hip_runtime.h>
#include <stdint.h>
#include <stddef.h>

// ---------------------------------------------------------------------------
// du_dt[i] = (1/n) * sum_{e : src[e]==i} weight[e] * sin(u[dst[e]] - u[src[e]])
//
// Memory-bound scatter kernel. Edge streams (weight/src/dst, 12B/edge, 384MB)
// are staged HBM -> LDS with gfx1250 async copies (double buffered, waited
// with s_wait_asynccnt), overlapping the streaming traffic with gather+sin+
// atomic work. u (2MB) and out (2MB) are L2-resident (192MB L2).
// sin computed with v_sin_f32 (__sinf): args are bounded (u ~ N(0,1)), and
// per-edge messages are scaled by 1/n = 2e-6, so the ~1e-6 abs error of the
// fast path vanishes relative to the atomic-order nondeterminism tolerance.
// ---------------------------------------------------------------------------

#define TPB   256                    // threads per block (8 waves of 32)
#define EPT   4                      // edges per thread per tile (one B128)
#define TILE  (TPB * EPT)            // 1024 edges per tile
#define ABYTES (TILE * 4)            // 4096 bytes per array per tile

// Async copy: 16 bytes from global (per lane) directly into LDS at the byte
// offset held in a VGPR. Tracked with ASYNCcnt. EXEC is uniform at all call
// sites (whole-block-uniform control flow).
__device__ __forceinline__ void async_b128(uint32_t lds_byte_off, const void* gaddr) {
    asm volatile("global_load_async_to_lds_b128 %0, %1, off"
                 :: "v"(lds_byte_off), "v"(gaddr)
                 : "memory");
}

#define S_WAIT_ASYNCCNT(n) asm volatile("s_wait_asynccnt " #n ::: "memory")

// Low 32 bits of a generic pointer into __shared__ = workgroup-relative LDS
// byte address (aperture lives in bits [63:32] per the flat-addressing spec);
// this is what the async instruction's VDST VGPR must hold.
__device__ __forceinline__ uint32_t lds_off32(const void* p) {
    return (uint32_t)(uintptr_t)p;
}

__global__ void zero_kernel(float* __restrict__ out, int n) {
    int i = blockIdx.x * blockDim.x + threadIdx.x;
    if (i < n) out[i] = 0.0f;
}

__global__ void __launch_bounds__(TPB)
edge_kernel(const float* __restrict__ u,
            const float* __restrict__ w,
            const int*   __restrict__ src,
            const int*   __restrict__ dst,
            float*       __restrict__ out,
            int n_tiles, float inv_n)
{
    // [buffer][array: 0=w 1=src 2=dst][bytes]
    __shared__ __align__(16) unsigned char smem[2][3][ABYTES];

    const int      t         = threadIdx.x;
    const uint32_t lane_byte = (uint32_t)t * 16u;

    uint32_t lw[2], ls[2], ld[2];
#pragma unroll
    for (int b = 0; b < 2; ++b) {
        lw[b] = lds_off32(&smem[b][0][lane_byte]);
        ls[b] = lds_off32(&smem[b][1][lane_byte]);
        ld[b] = lds_off32(&smem[b][2][lane_byte]);
    }

    const char* wB = (const char*)w;
    const char* sB = (const char*)src;
    const char* dB = (const char*)dst;

    const int stride = gridDim.x;
    int kt = blockIdx.x;
    if (kt >= n_tiles) return;

    // Prime the pipeline: issue tile kt into buffer 0.
    {
        size_t base = (size_t)kt * ABYTES + lane_byte;
        async_b128(lw[0], wB + base);
        async_b128(ls[0], sB + base);
        async_b128(ld[0], dB + base);
    }

    int buf = 0;
    for (; kt < n_tiles; kt += stride) {
        const int nxt = kt + stride;
        if (nxt < n_tiles) {
            // Issue the next tile into the other buffer, then wait until only
            // those 3 copies remain outstanding (async loads complete in
            // order, so the current tile's data is then resident in LDS).
            size_t base = (size_t)nxt * ABYTES + lane_byte;
            async_b128(lw[buf ^ 1], wB + base);
            async_b128(ls[buf ^ 1], sB + base);
            async_b128(ld[buf ^ 1], dB + base);
            S_WAIT_ASYNCCNT(3);
        } else {
            S_WAIT_ASYNCCNT(0);
        }

        // Each lane reads back exactly the 4 edges it staged (no cross-lane
        // sharing -> no barrier needed, only the per-wave asynccnt wait).
        const float4 wv = *reinterpret_cast<const float4*>(&smem[buf][0][lane_byte]);
        const int4   sv = *reinterpret_cast<const int4*>(&smem[buf][1][lane_byte]);
        const int4   dv = *reinterpret_cast<const int4*>(&smem[buf][2][lane_byte]);

        {
            // Fold 1/n into the weight, then one v_sin_f32 per edge.
            float m0 = (wv.x * inv_n) * __sinf(u[dv.x] - u[sv.x]);
            float m1 = (wv.y * inv_n) * __sinf(u[dv.y] - u[sv.y]);
            float m2 = (wv.z * inv_n) * __sinf(u[dv.z] - u[sv.z]);
            float m3 = (wv.w * inv_n) * __sinf(u[dv.w] - u[sv.w]);
            unsafeAtomicAdd(out + sv.x, m0);   // global_atomic_add_f32, no return
            unsafeAtomicAdd(out + sv.y, m1);
            unsafeAtomicAdd(out + sv.z, m2);
            unsafeAtomicAdd(out + sv.w, m3);
        }

        buf ^= 1;
    }
}

// Generic tail (n_edges not a multiple of TILE); plain coalesced path.
__global__ void tail_kernel(const float* __restrict__ u,
                            const float* __restrict__ w,
                            const int*   __restrict__ src,
                            const int*   __restrict__ dst,
                            float*       __restrict__ out,
                            int start, int n_edges, float inv_n)
{
    int e = start + blockIdx.x * blockDim.x + threadIdx.x;
    if (e < n_edges) {
        int s = src[e], d = dst[e];
        float m = (w[e] * inv_n) * __sinf(u[d] - u[s]);
        unsafeAtomicAdd(out + s, m);
    }
}

extern "C" void kernel_launch(void* const* d_in, const int* in_sizes, int n_in,
                              void* d_out, int out_size, void* d_ws, size_t ws_size,
                              hipStream_t stream) {
    // inputs: [0]=t (unused), [1]=u f32[N], [2]=weight f32[E], [3]=src i32[E], [4]=dst i32[E]
    const float* u   = (const float*)d_in[1];
    const float* w   = (const float*)d_in[2];
    const int*   src = (const int*)d_in[3];
    const int*   dst = (const int*)d_in[4];
    float*       out = (float*)d_out;

    const int n_nodes = in_sizes[1];
    const int n_edges = in_sizes[2];
    const float inv_n = 1.0f / (float)n_nodes;

    // Output accumulates via atomics: must start at zero every call.
    zero_kernel<<<(n_nodes + 255) / 256, 256, 0, stream>>>(out, n_nodes);

    const int n_tiles = n_edges / TILE;
    if (n_tiles > 0) {
        int blocks = n_tiles < 4096 ? n_tiles : 4096;
        edge_kernel<<<blocks, TPB, 0, stream>>>(u, w, src, dst, out, n_tiles, inv_n);
    }
    const int rem_start = n_tiles * TILE;
    const int rem = n_edges - rem_start;
    if (rem > 0) {
        tail_kernel<<<(rem + 255) / 256, 256, 0, stream>>>(u, w, src, dst, out,
                                                           rem_start, n_edges, inv_n);
    }
}